// My_Attention_7937099563537
// MI455X (gfx1250) — compile-verified
//
#include <hip/hip_runtime.h>
#include <math.h>

// Problem constants (from the reference)
#define BATCH 4096
#define SEQ   80
#define DIM   256
#define TOPICS 12
#define NROWS (BATCH * SEQ)          // 327680

typedef __attribute__((ext_vector_type(2))) float v2f;
typedef __attribute__((ext_vector_type(8))) float v8f;

// D = A(16x4 f32) * B(4x16 f32) + C(16x16 f32), full wave32 op.
__device__ __forceinline__ v8f wmma_f32(v2f a, v2f b, v8f c) {
    return __builtin_amdgcn_wmma_f32_16x16x4_f32(
        /*neg_a=*/false, a, /*neg_b=*/false, b,
        /*c_mod=*/(short)0, c, /*reuse_a=*/false, /*reuse_b=*/false);
}

// ---------------------------------------------------------------------------
// K1: scores[n, t] = sum_c x[n, c] * kernel[c, t]   via WMMA (A = kernel^T)
//     Also produces per-block max per topic (padded to 16) for the softmax.
// grid = NROWS/128 = 2560 blocks of 256 threads (8 waves, 16 rows per wave).
//
// Padding-lane trick: lanes with l16 >= 12 load A from a clamped in-bounds
// address (garbage value). Garbage in A rows M=12..15 only reaches D rows
// 12..15, which are never stored, and blockmax entries 12..15, which are
// never consumed downstream. No predicated loads needed.
// ---------------------------------------------------------------------------
__global__ void __launch_bounds__(256)
scores_kernel(const float* __restrict__ x, const float* __restrict__ kern,
              float* __restrict__ scores, float* __restrict__ blockmax)
{
    const int tid  = threadIdx.x;
    const int wave = tid >> 5;
    const int lane = tid & 31;
    const int l16  = lane & 15;           // A: topic m   /  B: row offset n
    const int kh   = lane >> 4;           // K half selector
    const int mt   = (l16 < TOPICS) ? l16 : (TOPICS - 1);  // clamped, in-bounds
    const long n0  = ((long)blockIdx.x * 8 + wave) * 16;

    __shared__ float smax[8][16];

    v8f acc = {};

    #pragma unroll
    for (int kk = 0; kk < 64; ++kk) {
        const int kbase = kk * 4 + 2 * kh;
        // B fragment: lane's own row, two consecutive columns (8B aligned).
        v2f b = *(const v2f*)(x + (n0 + l16) * DIM + kbase);
        // A fragment: kernel^T, unconditional loads from clamped column.
        v2f a;
        a.x = kern[kbase * TOPICS + mt];
        a.y = kern[(kbase + 1) * TOPICS + mt];
        acc = wmma_f32(a, b, acc);
    }

    // Store scores: D vgpr r -> topic t = r + 8*kh, row = n0 + l16.
    #pragma unroll
    for (int r = 0; r < 8; ++r) {
        const int t = r + 8 * kh;
        if (t < TOPICS) scores[(n0 + l16) * TOPICS + t] = acc[r];
    }

    // Per-topic max over this wave's 16 rows: reduce across the 16-lane half.
    #pragma unroll
    for (int r = 0; r < 8; ++r) {
        float v = acc[r];
        v = fmaxf(v, __shfl_xor(v, 1, 32));
        v = fmaxf(v, __shfl_xor(v, 2, 32));
        v = fmaxf(v, __shfl_xor(v, 4, 32));
        v = fmaxf(v, __shfl_xor(v, 8, 32));
        if (l16 == 0) smax[wave][8 * kh + r] = v;   // t >= 12 entries unused
    }
    __syncthreads();

    if (tid < 16) {
        float m = smax[0][tid];
        #pragma unroll
        for (int w = 1; w < 8; ++w) m = fmaxf(m, smax[w][tid]);
        blockmax[(long)blockIdx.x * 16 + tid] = m;
    }
}

// ---------------------------------------------------------------------------
// K2: global max per topic (single block, deterministic tree).
// ---------------------------------------------------------------------------
__global__ void __launch_bounds__(256)
reduce_max_kernel(const float* __restrict__ blockmax, float* __restrict__ gmax,
                  int nblocks)
{
    __shared__ float sred[256];
    const int tid = threadIdx.x;
    const int topic = tid & 15;
    const int chunk = tid >> 4;           // 16 strided chunks
    float m = -INFINITY;
    for (int i = chunk; i < nblocks; i += 16)
        m = fmaxf(m, blockmax[(long)i * 16 + topic]);
    sred[tid] = m;
    __syncthreads();
    for (int s = 128; s >= 16; s >>= 1) {
        if (tid < s) sred[tid] = fmaxf(sred[tid], sred[tid + s]);
        __syncthreads();
    }
    if (tid < 16) gmax[tid] = sred[tid];
}

// ---------------------------------------------------------------------------
// K3: wun[n,t] = exp(scores[n,t] - gmax[t]); per-block partial sums (pad 16).
// grid = NROWS/256 = 1280 blocks; one row per thread.
// ---------------------------------------------------------------------------
__global__ void __launch_bounds__(256)
expsum_kernel(const float* __restrict__ scores, const float* __restrict__ gmax,
              float* __restrict__ wun, float* __restrict__ blocksum)
{
    __shared__ float ssum[TOPICS * 256];
    const int tid = threadIdx.x;
    const long n = (long)blockIdx.x * 256 + tid;

    float loc[TOPICS];
    #pragma unroll
    for (int t = 0; t < TOPICS; ++t) {
        float e = __expf(scores[n * TOPICS + t] - gmax[t]);
        wun[n * TOPICS + t] = e;
        loc[t] = e;
    }
    #pragma unroll
    for (int t = 0; t < TOPICS; ++t) ssum[t * 256 + tid] = loc[t];
    __syncthreads();
    for (int s = 128; s > 0; s >>= 1) {
        if (tid < s) {
            #pragma unroll
            for (int t = 0; t < TOPICS; ++t)
                ssum[t * 256 + tid] += ssum[t * 256 + tid + s];
        }
        __syncthreads();
    }
    if (tid < 16)
        blocksum[(long)blockIdx.x * 16 + tid] = (tid < TOPICS) ? ssum[tid * 256] : 0.0f;
}

// ---------------------------------------------------------------------------
// K4: global sums -> ginv[16] (1/sum for t<12, 0 for padding topics).
// ---------------------------------------------------------------------------
__global__ void __launch_bounds__(256)
reduce_sum_kernel(const float* __restrict__ blocksum, float* __restrict__ ginv,
                  int nblocks)
{
    __shared__ float sred[256];
    const int tid = threadIdx.x;
    const int topic = tid & 15;
    const int chunk = tid >> 4;
    float s = 0.0f;
    for (int i = chunk; i < nblocks; i += 16)
        s += blocksum[(long)i * 16 + topic];
    sred[tid] = s;
    __syncthreads();
    for (int st = 128; st >= 16; st >>= 1) {
        if (tid < st) sred[tid] += sred[tid + st];
        __syncthreads();
    }
    if (tid < 16) {
        float tot = sred[tid];
        ginv[tid] = (tid < TOPICS && tot != 0.0f) ? (1.0f / tot) : 0.0f;
    }
}

// ---------------------------------------------------------------------------
// K5: vs[t, b2, c] = sum_{i<80} W[i*B + b2, t] * x[i*B + b2, c]
// One wave per b2. A = W^T (16x80, 12 real topics) held in 40 VGPRs,
// 16 column tiles x 20 K-steps of v_wmma_f32_16x16x4_f32.
// grid = BATCH/8 = 512 blocks of 256 threads.
//
// Padding lanes load A from a clamped in-bounds address and are zeroed by
// inv_m == 0 (ginv is zero-padded) -> unconditional loads, no exec branching.
// ---------------------------------------------------------------------------
__global__ void __launch_bounds__(256)
einsum_kernel(const float* __restrict__ x, const float* __restrict__ wun,
              const float* __restrict__ ginv, float* __restrict__ out)
{
    const int tid  = threadIdx.x;
    const int wave = tid >> 5;
    const int lane = tid & 31;
    const int l16  = lane & 15;           // A: topic m  /  B,D: column offset
    const int kh   = lane >> 4;
    const int mt   = (l16 < TOPICS) ? l16 : (TOPICS - 1);  // clamped address
    const long b2  = (long)blockIdx.x * 8 + wave;

    const float inv_m = ginv[l16];        // padded: 0 for l16 >= 12

    // A fragments for all K=80 (kept resident in VGPRs).
    v2f areg[20];
    #pragma unroll
    for (int kk = 0; kk < 20; ++kk) {
        const int i0 = kk * 4 + 2 * kh;   // sequence index = K index
        float a0 = wun[((long)i0 * BATCH + b2) * TOPICS + mt];
        float a1 = wun[((long)(i0 + 1) * BATCH + b2) * TOPICS + mt];
        v2f a; a.x = a0 * inv_m; a.y = a1 * inv_m;
        areg[kk] = a;
    }

    for (int ct = 0; ct < 16; ++ct) {
        const int c0 = ct * 16;
        v8f acc = {};
        #pragma unroll
        for (int kk = 0; kk < 20; ++kk) {
            const int i0 = kk * 4 + 2 * kh;
            // B fragment: rows i0, i0+1 of the stride-B regrouping; 64B
            // contiguous per half-wave, lines reused across the 16 c-tiles.
            v2f b;
            b.x = x[((long)i0 * BATCH + b2) * DIM + c0 + l16];
            b.y = x[((long)(i0 + 1) * BATCH + b2) * DIM + c0 + l16];
            acc = wmma_f32(areg[kk], b, acc);
        }
        #pragma unroll
        for (int r = 0; r < 8; ++r) {
            const int t = r + 8 * kh;
            if (t < TOPICS)
                out[((long)t * BATCH + b2) * DIM + c0 + l16] = acc[r];
        }
    }
}

// ---------------------------------------------------------------------------
extern "C" void kernel_launch(void* const* d_in, const int* in_sizes, int n_in,
                              void* d_out, int out_size, void* d_ws, size_t ws_size,
                              hipStream_t stream) {
    const float* x    = (const float*)d_in[0];   // (4096, 80, 256) f32
    const float* kern = (const float*)d_in[1];   // (256, 12) f32
    float* out = (float*)d_out;                  // (12, 4096, 256) f32

    // Workspace layout (floats): ~31.7 MB total.
    float* scores   = (float*)d_ws;                       // NROWS*12
    float* wun      = scores   + (long)NROWS * TOPICS;    // NROWS*12
    float* blockmax = wun      + (long)NROWS * TOPICS;    // 2560*16
    float* blocksum = blockmax + 2560 * 16;               // 1280*16
    float* gmax     = blocksum + 1280 * 16;               // 16
    float* ginv     = gmax     + 16;                      // 16

    scores_kernel    <<<NROWS / 128, 256, 0, stream>>>(x, kern, scores, blockmax);
    reduce_max_kernel<<<1,           256, 0, stream>>>(blockmax, gmax, NROWS / 128);
    expsum_kernel    <<<NROWS / 256, 256, 0, stream>>>(scores, gmax, wun, blocksum);
    reduce_sum_kernel<<<1,           256, 0, stream>>>(blocksum, ginv, NROWS / 256);
    einsum_kernel    <<<BATCH / 8,   256, 0, stream>>>(x, wun, ginv, out);
}